// ResearchTGN_47725676593774
// MI455X (gfx1250) — compile-verified
//
#include <hip/hip_runtime.h>
#include <hip/hip_bf16.h>

// ---------------------------------------------------------------------------
// TGN: 2x TransformerConv (H=4, C=32, D=128), fp32, WMMA f32_16x16x4 GEMMs.
// ---------------------------------------------------------------------------

typedef __attribute__((ext_vector_type(2))) float v2f;
typedef __attribute__((ext_vector_type(8))) float v8f;

#define D_DIM   128
#define HC      128
#define NHEAD   4
#define CH      32
#define PE_RAW  73
#define PE_PAD  80
#define SCALE_F 0.17677669529663687f  // 1/sqrt(32)

#define BM 128
#define BN 64
#define BK 16
#define LP 20   // LDS row pitch: 16B-aligned rows, conflict-free column reads

// ---- monotonic float<->uint encoding for atomic segment-max ---------------
__device__ __forceinline__ unsigned ord_encode(float f) {
  unsigned u = __float_as_uint(f);
  return (u & 0x80000000u) ? ~u : (u ^ 0x80000000u);
}
__device__ __forceinline__ float ord_decode(unsigned u) {
  unsigned b = (u & 0x80000000u) ? (u ^ 0x80000000u) : ~u;
  return __uint_as_float(b);
}

// ---------------------------------------------------------------------------
// WMMA fp32 GEMM: C[M,N] = A[M,K] @ B[K,N] (+ bias[N]).
// Block = 128x64 tile, 256 threads = 8 wave32 stacked in M; each wave owns a
// 16x64 strip: one A fragment reused across 4 B fragments -> 4 WMMAs per
// k-step. B is staged TRANSPOSED in LDS so each fragment is one ds_load_b64.
// Requires: N % 64 == 0, K % 16 == 0 (all call sites: K in {80,128}, N = 128).
// M is guarded (clamped loads, guarded stores).
// ---------------------------------------------------------------------------
__global__ __launch_bounds__(256) void wmma_gemm_f32(
    const float* __restrict__ A, const float* __restrict__ B,
    const float* __restrict__ bias, float* __restrict__ C,
    int M, int N, int K)
{
  __shared__ float As[BM][LP];   // 128 x 20 (row-major, K contiguous)
  __shared__ float Bt[BN][LP];   // 64 x 20  (transposed: [col][k])

  const int tid  = threadIdx.x;
  const int lane = tid & 31;
  const int wave = tid >> 5;        // 0..7
  const int wm   = wave << 4;       // wave's 16-row strip
  const int bm   = blockIdx.x * BM;
  const int bn   = blockIdx.y * BN;

  const int mn   = lane & 15;         // A row / B col within 16x16
  const int koff = (lane >> 4) << 1;  // lanes 0-15 -> K{0,1}, 16-31 -> K{2,3}

  v8f acc[4] = {{}, {}, {}, {}};

  // A tile: 128x16 floats; 256 threads x 8 floats (two float4)
  const int ar   = tid >> 1;          // 0..127
  const int ac   = (tid & 1) << 3;    // 0 or 8
  const int arow = (bm + ar < M) ? (bm + ar) : (M - 1);
  // B tile: 16x64 floats; 256 threads x 1 float4, stored transposed
  const int br = tid >> 4;            // 0..15 (K row)
  const int bc = (tid & 15) << 2;     // 0..60 (col)

  for (int k0 = 0; k0 < K; k0 += BK) {
    float4 a0 = *(const float4*)(A + (size_t)arow * K + k0 + ac);
    float4 a1 = *(const float4*)(A + (size_t)arow * K + k0 + ac + 4);
    float4 bv = *(const float4*)(B + (size_t)(k0 + br) * N + bn + bc);
    *(float4*)&As[ar][ac]     = a0;
    *(float4*)&As[ar][ac + 4] = a1;
    Bt[bc + 0][br] = bv.x;
    Bt[bc + 1][br] = bv.y;
    Bt[bc + 2][br] = bv.z;
    Bt[bc + 3][br] = bv.w;
    __syncthreads();

#pragma unroll
    for (int kk = 0; kk < BK; kk += 4) {
      v2f a = *(const v2f*)&As[wm + mn][kk + koff];
#pragma unroll
      for (int nt = 0; nt < 4; ++nt) {
        v2f b = *(const v2f*)&Bt[(nt << 4) + mn][kk + koff];
        acc[nt] = __builtin_amdgcn_wmma_f32_16x16x4_f32(
            false, a, false, b, (short)0, acc[nt], false, false);
      }
    }
    __syncthreads();
  }

  // C/D layout: VGPR g -> row (g + 8*(lane>=16)), col = lane&15
  const int rbase = bm + wm + ((lane >> 4) << 3);
#pragma unroll
  for (int nt = 0; nt < 4; ++nt) {
    const int c = bn + (nt << 4) + mn;
    const float bb = bias ? bias[c] : 0.0f;
#pragma unroll
    for (int g = 0; g < 8; ++g) {
      int r = rbase + g;
      if (r < M) C[(size_t)r * N + c] = acc[nt][g] + bb;
    }
  }
}

// ---------------------------------------------------------------------------
// x[n, :] = memory_table[n_id[n], :]   (float4 per thread)
// ---------------------------------------------------------------------------
__global__ __launch_bounds__(256) void k_gather(
    const float* __restrict__ mt, const int* __restrict__ n_id,
    float* __restrict__ x, int N)
{
  int i = blockIdx.x * blockDim.x + threadIdx.x;     // over N*32 float4s
  if (i >= N * 32) return;
  int n = i >> 5;
  int c = i & 31;
  const float4* s = (const float4*)(mt + (size_t)n_id[n] * D_DIM) + c;
  ((float4*)(x + (size_t)n * D_DIM))[c] = *s;
}

// ---------------------------------------------------------------------------
// pe[e] = concat(edge_attr[e,0:3], edge_attr[e,4:66], et_emb[etype]) zero-pad to 80
// ---------------------------------------------------------------------------
__global__ __launch_bounds__(256) void k_build_pe(
    const float* __restrict__ ea, const float* __restrict__ et_emb,
    float* __restrict__ pe, int E)
{
  int e = blockIdx.x * blockDim.x + threadIdx.x;
  if (e >= E) return;
  const float* row = ea + (size_t)e * 66;
  float* out = pe + (size_t)e * PE_PAD;
  out[0] = row[0]; out[1] = row[1]; out[2] = row[2];
#pragma unroll
  for (int j = 0; j < 62; ++j) out[3 + j] = row[4 + j];
  int et = (int)row[3];
#pragma unroll
  for (int j = 0; j < 8; ++j) out[65 + j] = et_emb[et * 8 + j];
#pragma unroll
  for (int j = PE_RAW; j < PE_PAD; ++j) out[j] = 0.0f;
}

// We [73,128] -> zero-padded [80,128]
__global__ __launch_bounds__(256) void k_pad_w(
    const float* __restrict__ We, float* __restrict__ Wp)
{
  int i = blockIdx.x * blockDim.x + threadIdx.x;
  if (i >= PE_PAD * HC) return;
  int r = i >> 7;
  Wp[i] = (r < PE_RAW) ? We[i] : 0.0f;
}

// m = encode(-inf), denom = 0, agg = 0
__global__ __launch_bounds__(256) void k_init(
    unsigned* __restrict__ m, float* __restrict__ denom,
    float* __restrict__ agg, int N)
{
  int i = blockIdx.x * blockDim.x + threadIdx.x;
  if (i < N * NHEAD) { m[i] = 0x007FFFFFu; denom[i] = 0.0f; }  // encode(-inf)
  if (i < N * HC) agg[i] = 0.0f;
}

// per (edge, head): score = q[dst]·(k[src]+e) * SCALE; atomic segment max
__global__ __launch_bounds__(256) void k_scores(
    const float* __restrict__ q, const float* __restrict__ k,
    const float* __restrict__ e, const int* __restrict__ src,
    const int* __restrict__ dst, float* __restrict__ scores,
    unsigned* __restrict__ m, int E)
{
  int idx = blockIdx.x * blockDim.x + threadIdx.x;
  if (idx >= E * NHEAD) return;
  int eid = idx >> 2, h = idx & 3;
  int s = src[eid], d = dst[eid];
  const float* qp = q + (size_t)d * HC + h * CH;
  const float* kp = k + (size_t)s * HC + h * CH;
  const float* ep = e + (size_t)eid * HC + h * CH;
  float acc = 0.0f;
#pragma unroll
  for (int c = 0; c < CH; c += 4) {
    float4 qv = *(const float4*)(qp + c);
    float4 kv = *(const float4*)(kp + c);
    float4 ev = *(const float4*)(ep + c);
    acc += qv.x * (kv.x + ev.x) + qv.y * (kv.y + ev.y)
         + qv.z * (kv.z + ev.z) + qv.w * (kv.w + ev.w);
  }
  float sc = acc * SCALE_F;
  scores[idx] = sc;
  atomicMax(&m[(size_t)d * NHEAD + h], ord_encode(sc));
}

// per (edge, head): ex = exp(score - m); denom += ex; agg += ex*(v[src]+e)
__global__ __launch_bounds__(256) void k_accum(
    const float* __restrict__ v, const float* __restrict__ e,
    const int* __restrict__ src, const int* __restrict__ dst,
    const float* __restrict__ scores, const unsigned* __restrict__ m,
    float* __restrict__ denom, float* __restrict__ agg, int E)
{
  int idx = blockIdx.x * blockDim.x + threadIdx.x;
  if (idx >= E * NHEAD) return;
  int eid = idx >> 2, h = idx & 3;
  int s = src[eid], d = dst[eid];
  float mx = ord_decode(m[(size_t)d * NHEAD + h]);
  float ex = expf(scores[idx] - mx);
  atomicAdd(&denom[(size_t)d * NHEAD + h], ex);
  const float* vp = v + (size_t)s * HC + h * CH;
  const float* ep = e + (size_t)eid * HC + h * CH;
  float* ap = agg + (size_t)d * HC + h * CH;
#pragma unroll
  for (int c = 0; c < CH; ++c) atomicAdd(ap + c, ex * (vp[c] + ep[c]));
}

// out = agg/(denom+1e-16) + skip; optional relu
__global__ __launch_bounds__(256) void k_finalize(
    const float* __restrict__ agg, const float* __restrict__ denom,
    const float* __restrict__ skip, float* __restrict__ out, int N, int relu)
{
  int i = blockIdx.x * blockDim.x + threadIdx.x;
  if (i >= N * HC) return;
  int n = i >> 7;
  int h = (i >> 5) & 3;
  float o = agg[i] / (denom[n * NHEAD + h] + 1e-16f) + skip[i];
  if (relu) o = fmaxf(o, 0.0f);
  out[i] = o;
}

// ---------------------------------------------------------------------------
extern "C" void kernel_launch(void* const* d_in, const int* in_sizes, int n_in,
                              void* d_out, int out_size, void* d_ws, size_t ws_size,
                              hipStream_t stream) {
  const float* memory_table = (const float*)d_in[0];
  const float* edge_attr    = (const float*)d_in[1];
  const float* et_emb       = (const float*)d_in[2];
  const float* Wq[2] = {(const float*)d_in[3],  (const float*)d_in[12]};
  const float* Wk[2] = {(const float*)d_in[4],  (const float*)d_in[13]};
  const float* Wv[2] = {(const float*)d_in[5],  (const float*)d_in[14]};
  const float* We[2] = {(const float*)d_in[6],  (const float*)d_in[15]};
  const float* Ws[2] = {(const float*)d_in[7],  (const float*)d_in[16]};
  const float* bq[2] = {(const float*)d_in[8],  (const float*)d_in[17]};
  const float* bk[2] = {(const float*)d_in[9],  (const float*)d_in[18]};
  const float* bv[2] = {(const float*)d_in[10], (const float*)d_in[19]};
  const float* bs[2] = {(const float*)d_in[11], (const float*)d_in[20]};
  const int*   n_id = (const int*)d_in[21];
  const int*   src  = (const int*)d_in[22];

  const int N = in_sizes[21];        // 40000
  const int E = in_sizes[22] / 2;    // 320000
  const int* dst = src + E;

  // ---- workspace carve-out -------------------------------------------------
  char* ws = (char*)d_ws;
  size_t off = 0;
  auto carve = [&](size_t bytes) {
    void* p = ws + off;
    off += (bytes + 255) & ~(size_t)255;
    return p;
  };
  float*    pe     = (float*)   carve((size_t)E * PE_PAD * 4);
  float*    ebuf   = (float*)   carve((size_t)E * HC * 4);
  float*    x      = (float*)   carve((size_t)N * HC * 4);
  float*    x2     = (float*)   carve((size_t)N * HC * 4);
  float*    qb     = (float*)   carve((size_t)N * HC * 4);
  float*    kb     = (float*)   carve((size_t)N * HC * 4);
  float*    vb     = (float*)   carve((size_t)N * HC * 4);
  float*    sb     = (float*)   carve((size_t)N * HC * 4);
  float*    scores = (float*)   carve((size_t)E * NHEAD * 4);
  unsigned* mmax   = (unsigned*)carve((size_t)N * NHEAD * 4);
  float*    denom  = (float*)   carve((size_t)N * NHEAD * 4);
  float*    agg    = (float*)   carve((size_t)N * HC * 4);
  float*    Wp[2];
  Wp[0] = (float*)carve((size_t)PE_PAD * HC * 4);
  Wp[1] = (float*)carve((size_t)PE_PAD * HC * 4);

  auto gemm = [&](const float* A, const float* B, const float* bias, float* C,
                  int M, int Ncols, int K) {
    dim3 grid((M + BM - 1) / BM, (Ncols + BN - 1) / BN);
    wmma_gemm_f32<<<grid, dim3(256), 0, stream>>>(A, B, bias, C, M, Ncols, K);
  };

  // ---- shared preprocessing ------------------------------------------------
  k_gather<<<(N * 32 + 255) / 256, 256, 0, stream>>>(memory_table, n_id, x, N);
  k_build_pe<<<(E + 255) / 256, 256, 0, stream>>>(edge_attr, et_emb, pe, E);
  k_pad_w<<<(PE_PAD * HC + 255) / 256, 256, 0, stream>>>(We[0], Wp[0]);
  k_pad_w<<<(PE_PAD * HC + 255) / 256, 256, 0, stream>>>(We[1], Wp[1]);

  // ---- two TransformerConv layers -----------------------------------------
  for (int layer = 0; layer < 2; ++layer) {
    const float* xin = (layer == 0) ? x : x2;
    float* out = (layer == 0) ? x2 : (float*)d_out;

    gemm(pe,  Wp[layer], nullptr,   ebuf, E, HC, PE_PAD);  // e  = pe @ We
    gemm(xin, Wq[layer], bq[layer], qb,   N, HC, D_DIM);   // q  = x @ Wq + bq
    gemm(xin, Wk[layer], bk[layer], kb,   N, HC, D_DIM);   // k  = x @ Wk + bk
    gemm(xin, Wv[layer], bv[layer], vb,   N, HC, D_DIM);   // v  = x @ Wv + bv
    gemm(xin, Ws[layer], bs[layer], sb,   N, HC, D_DIM);   // skip

    k_init<<<(N * HC + 255) / 256, 256, 0, stream>>>(mmax, denom, agg, N);
    k_scores<<<(E * NHEAD + 255) / 256, 256, 0, stream>>>(
        qb, kb, ebuf, src, dst, scores, mmax, E);
    k_accum<<<(E * NHEAD + 255) / 256, 256, 0, stream>>>(
        vb, ebuf, src, dst, scores, mmax, denom, agg, E);
    k_finalize<<<(N * HC + 255) / 256, 256, 0, stream>>>(
        agg, denom, sb, out, N, layer == 0 ? 1 : 0);
  }
}